// JpegLayer_65996467470859
// MI455X (gfx1250) — compile-verified
//
#include <hip/hip_runtime.h>
#include <math.h>

// ---------------------------------------------------------------------------
// Fused differentiable-JPEG forward for MI455X (gfx1250, wave32).
//   - one global read + one global write per pixel (memory-bound: ~201 MB
//     -> ~8.6 us floor at 23.3 TB/s), non-temporal streaming hints
//   - 8x8 DCT/IDCT as 16x16 block-diagonal sandwiches via
//     V_WMMA_F32_16X16X4_F32 (16 WMMAs per 16x16 tile, 96 per wave)
//   - wave-private LDS staging; s_wait_dscnt only where a cross-lane
//     write->read hazard exists (DS ops of a wave execute in order, so
//     read-then-overwrite needs no fence)
//   - quantizer uses precomputed reciprocals (no fp32 divides in the loop)
// ---------------------------------------------------------------------------

typedef float v2f __attribute__((ext_vector_type(2)));
typedef float v8f __attribute__((ext_vector_type(8)));

#define LEVEL (128.0f / 255.0f)

// Per-wave LDS region: Y 32x32 (stride 33) + Cb/Cr 16x16 (stride 17) + scratch
#define Y_STRIDE   33
#define C_STRIDE   17
#define Y_WORDS    (32 * Y_STRIDE)          // 1056
#define C_WORDS    (16 * C_STRIDE)          // 272
#define WAVE_WORDS (Y_WORDS + 3 * C_WORDS)  // 1872 floats = 7488 B
#define NWAVES     8

__device__ __forceinline__ v8f wmma_f32(v2f a, v2f b, v8f c) {
  // D = A(16x4) * B(4x16) + C ; 8 args: neg_a, A, neg_b, B, c_mod, C, reuse_a, reuse_b
  return __builtin_amdgcn_wmma_f32_16x16x4_f32(false, a, false, b, (short)0, c,
                                               false, false);
}

// Wave-local LDS ordering for cross-lane write->read hazards. LDS regions are
// wave-private, so s_wait_dscnt 0 suffices (no workgroup barrier).
__device__ __forceinline__ void lds_fence() {
  __builtin_amdgcn_wave_barrier();
  asm volatile("s_wait_dscnt 0" ::: "memory");
  __builtin_amdgcn_wave_barrier();
}

// Entry (i,j) of M2 = blockdiag(DCT8, DCT8), DCT8 = type-II orthonormal 8-pt DCT.
__device__ __forceinline__ float m2v(int i, int j) {
  if ((i >> 3) != (j >> 3)) return 0.0f;
  int ii = i & 7, jj = j & 7;
  if (ii == 0) return 0.35355339059327373f;  // 1/sqrt(8)
  return 0.5f * cosf(3.14159265358979323846f * (float)((2 * jj + 1) * ii) *
                     (1.0f / 16.0f));
}

__device__ __forceinline__ void rgb2ycc(float r, float g, float b, float& y,
                                        float& cb, float& cr) {
  y  = 0.299f * r + 0.587f * g + 0.114f * b;
  cb = -0.168735892f * r - 0.331264108f * g + 0.5f * b;
  cr = 0.5f * r - 0.418687589f * g - 0.081312411f * b;
  y  = fminf(fmaxf(y, 0.0f), 1.0f) - LEVEL;
  cb = fminf(fmaxf(cb + LEVEL, 0.0f), 1.0f) - LEVEL;
  cr = fminf(fmaxf(cr + LEVEL, 0.0f), 1.0f) - LEVEL;
}

__device__ __forceinline__ void ycc2rgb(float y, float cb, float cr, float& r,
                                        float& g, float& b) {
  float yc = y + LEVEL;
  r = yc + 1.402f * cr;
  g = yc - 0.344136286f * cb - 0.714136286f * cr;
  b = yc + 1.772f * cb;
  r = fminf(fmaxf(r, 0.0f), 1.0f);
  g = fminf(fmaxf(g, 0.0f), 1.0f);
  b = fminf(fmaxf(b, 0.0f), 1.0f);
}

// WMMA fragment layouts (CDNA5 ISA, f32 16x16):
//   A 16x4 : lane L holds row m=L&15, k = k0 + 2*(L>>4) + vgpr
//   B 4x16 : lane L holds col n=L&15, k = k0 + 2*(L>>4) + vgpr
//   C/D    : lane L holds col n=L&15, row m = vgpr + 8*(L>>4)
//
// codec_tile: T(16x16, stride st, LDS) -> DCT -> quant/dequant -> IDCT, in
// place. Caller must fence once after all tiles before consuming T.
__device__ __forceinline__ void codec_tile(float* T, int st, float* S,
                                           const v2f* crow, const v2f* ccol,
                                           const float* qv, const float* qrv,
                                           int ln, int hl) {
  // ---- Q = T * M2^T ----  (A from T, B[j] = M2t[k][n] = M2[n][k] = crow)
  v8f acc = {0.f, 0.f, 0.f, 0.f, 0.f, 0.f, 0.f, 0.f};
#pragma unroll
  for (int s = 0; s < 4; ++s) {
    int kb = 4 * s + 2 * hl;
    v2f a;
    a.x = T[ln * st + kb];
    a.y = T[ln * st + kb + 1];
    acc = wmma_f32(a, crow[s], acc);
  }
#pragma unroll
  for (int v = 0; v < 8; ++v) S[(v + 8 * hl) * C_STRIDE + ln] = acc[v];
  lds_fence();  // cross-lane write->read

  // ---- D = M2 * Q ----  (A[j] = M2[m][k] = crow, B from S)
  v8f d = {0.f, 0.f, 0.f, 0.f, 0.f, 0.f, 0.f, 0.f};
#pragma unroll
  for (int s = 0; s < 4; ++s) {
    int kb = 4 * s + 2 * hl;
    v2f b;
    b.x = S[kb * C_STRIDE + ln];
    b.y = S[(kb + 1) * C_STRIDE + ln];
    d = wmma_f32(crow[s], b, d);
  }
  // quantize / dequantize in C/D layout: element (m = v+8*hl, n = ln),
  // qtable index (m&7, n&7) = (v, ln&7) -> per-lane column qv[v].
#pragma unroll
  for (int v = 0; v < 8; ++v) d[v] = rintf(d[v] * qrv[v]) * qv[v];
  // DS ops execute in order within a wave: product-2 loads precede these
  // stores, so no fence needed before overwriting S.
#pragma unroll
  for (int v = 0; v < 8; ++v) S[(v + 8 * hl) * C_STRIDE + ln] = d[v];
  lds_fence();  // cross-lane write->read

  // ---- Q2 = dec * M2 ----  (A from S, B[j] = M2[k][n] = ccol)
  v8f p = {0.f, 0.f, 0.f, 0.f, 0.f, 0.f, 0.f, 0.f};
#pragma unroll
  for (int s = 0; s < 4; ++s) {
    int kb = 4 * s + 2 * hl;
    v2f a;
    a.x = S[ln * C_STRIDE + kb];
    a.y = S[ln * C_STRIDE + kb + 1];
    p = wmma_f32(a, ccol[s], p);
  }
#pragma unroll
  for (int v = 0; v < 8; ++v) S[(v + 8 * hl) * C_STRIDE + ln] = p[v];
  lds_fence();  // cross-lane write->read

  // ---- out = M2^T * Q2 ----  (A[j] = M2t[m][k] = M2[k][m] = ccol, B from S)
  v8f o = {0.f, 0.f, 0.f, 0.f, 0.f, 0.f, 0.f, 0.f};
#pragma unroll
  for (int s = 0; s < 4; ++s) {
    int kb = 4 * s + 2 * hl;
    v2f b;
    b.x = S[kb * C_STRIDE + ln];
    b.y = S[(kb + 1) * C_STRIDE + ln];
    o = wmma_f32(ccol[s], b, o);
  }
#pragma unroll
  for (int v = 0; v < 8; ++v) T[(v + 8 * hl) * st + ln] = o[v];
  // no trailing fence: next tile only writes S (ordered after our S reads);
  // caller fences once before phase 3 reads T.
}

__global__ __launch_bounds__(256) void jpeg_fused_kernel(
    const float* __restrict__ in, const float* __restrict__ quant,
    float* __restrict__ out) {
  __shared__ float smem[NWAVES * WAVE_WORDS];

  const int tid = threadIdx.x;
  const int wid = tid >> 5;
  const int lane = tid & 31;
  const int ln = lane & 15;
  const int hl = lane >> 4;

  float* yT = smem + wid * WAVE_WORDS;
  float* cbT = yT + Y_WORDS;
  float* crT = cbT + C_WORDS;
  float* S = crT + C_WORDS;

  // Per-lane constant WMMA fragments of M2 / M2^T.
  v2f crow[4], ccol[4];
#pragma unroll
  for (int s = 0; s < 4; ++s) {
    int kb = 4 * s + 2 * hl;
    crow[s].x = m2v(ln, kb);
    crow[s].y = m2v(ln, kb + 1);
    ccol[s].x = m2v(kb, ln);
    ccol[s].y = m2v(kb + 1, ln);
  }
  // Reference uses only quantize[0] for all three planes. Precompute the
  // per-lane qtable column and its reciprocal (quantizer becomes mul+rndne+mul).
  float qv[8], qrv[8];
#pragma unroll
  for (int v = 0; v < 8; ++v) {
    qv[v] = rintf(quant[v * 8 + (ln & 7)] * 255.0f) * (1.0f / 255.0f);
    qrv[v] = 1.0f / qv[v];
  }

  // One wave <-> one 32x32 pixel region. 32 images * 16 * 16 = 8192 regions.
  const int region = blockIdx.x * NWAVES + wid;
  const int b = region >> 8;
  const int rr = region & 255;
  const int rh = (rr >> 4) << 5;
  const int rw = (rr & 15) << 5;

  const float* inR = in + (size_t)b * 3 * 512 * 512;
  const float* inG = inR + 512 * 512;
  const float* inB = inG + 512 * 512;
  float* outR = out + (size_t)b * 3 * 512 * 512;
  float* outG = outR + 512 * 512;
  float* outB = outG + 512 * 512;

  // ---- Phase 1: load RGB (streaming/NT), RGB->YCC, 2x2 chroma average ----
  // Each lane owns one 2x2 quad per iteration (256 quads / 32 lanes = 8 iters).
#pragma unroll
  for (int it = 0; it < 8; ++it) {
    int q = it * 32 + lane;
    int qh = q >> 4, qw = q & 15;
    int h = rh + qh * 2, w = rw + qw * 2;
    size_t o0 = (size_t)h * 512 + w;
    v2f r0 = __builtin_nontemporal_load((const v2f*)(inR + o0));
    v2f r1 = __builtin_nontemporal_load((const v2f*)(inR + o0 + 512));
    v2f g0 = __builtin_nontemporal_load((const v2f*)(inG + o0));
    v2f g1 = __builtin_nontemporal_load((const v2f*)(inG + o0 + 512));
    v2f b0 = __builtin_nontemporal_load((const v2f*)(inB + o0));
    v2f b1 = __builtin_nontemporal_load((const v2f*)(inB + o0 + 512));

    float y00, y01, y10, y11, cb0, cb1, cb2, cb3, cr0, cr1, cr2, cr3;
    rgb2ycc(r0.x, g0.x, b0.x, y00, cb0, cr0);
    rgb2ycc(r0.y, g0.y, b0.y, y01, cb1, cr1);
    rgb2ycc(r1.x, g1.x, b1.x, y10, cb2, cr2);
    rgb2ycc(r1.y, g1.y, b1.y, y11, cb3, cr3);

    int yb = (2 * qh) * Y_STRIDE + 2 * qw;
    yT[yb] = y00;
    yT[yb + 1] = y01;
    yT[yb + Y_STRIDE] = y10;
    yT[yb + Y_STRIDE + 1] = y11;
    cbT[qh * C_STRIDE + qw] = 0.25f * (cb0 + cb1 + cb2 + cb3);
    crT[qh * C_STRIDE + qw] = 0.25f * (cr0 + cr1 + cr2 + cr3);
  }
  lds_fence();

  // ---- Phase 2: per-tile DCT -> quant -> IDCT (96 WMMAs / wave) ----
#pragma unroll
  for (int ty = 0; ty < 2; ++ty)
#pragma unroll
    for (int tx = 0; tx < 2; ++tx)
      codec_tile(yT + (ty * 16) * Y_STRIDE + tx * 16, Y_STRIDE, S, crow, ccol,
                 qv, qrv, ln, hl);
  codec_tile(cbT, C_STRIDE, S, crow, ccol, qv, qrv, ln, hl);
  codec_tile(crT, C_STRIDE, S, crow, ccol, qv, qrv, ln, hl);
  lds_fence();  // make all reconstructed tiles visible across lanes

  // ---- Phase 3: chroma upsample, YCC->RGB, clamp, store (streaming/NT) ----
#pragma unroll
  for (int it = 0; it < 8; ++it) {
    int q = it * 32 + lane;
    int qh = q >> 4, qw = q & 15;
    int h = rh + qh * 2, w = rw + qw * 2;
    size_t o0 = (size_t)h * 512 + w;

    int yb = (2 * qh) * Y_STRIDE + 2 * qw;
    float y00 = yT[yb], y01 = yT[yb + 1];
    float y10 = yT[yb + Y_STRIDE], y11 = yT[yb + Y_STRIDE + 1];
    float cb = cbT[qh * C_STRIDE + qw];
    float cr = crT[qh * C_STRIDE + qw];

    float r00, g00, b00, r01, g01, b01, r10, g10, b10, r11, g11, b11;
    ycc2rgb(y00, cb, cr, r00, g00, b00);
    ycc2rgb(y01, cb, cr, r01, g01, b01);
    ycc2rgb(y10, cb, cr, r10, g10, b10);
    ycc2rgb(y11, cb, cr, r11, g11, b11);

    v2f t;
    t.x = r00; t.y = r01; __builtin_nontemporal_store(t, (v2f*)(outR + o0));
    t.x = r10; t.y = r11; __builtin_nontemporal_store(t, (v2f*)(outR + o0 + 512));
    t.x = g00; t.y = g01; __builtin_nontemporal_store(t, (v2f*)(outG + o0));
    t.x = g10; t.y = g11; __builtin_nontemporal_store(t, (v2f*)(outG + o0 + 512));
    t.x = b00; t.y = b01; __builtin_nontemporal_store(t, (v2f*)(outB + o0));
    t.x = b10; t.y = b11; __builtin_nontemporal_store(t, (v2f*)(outB + o0 + 512));
  }
}

extern "C" void kernel_launch(void* const* d_in, const int* in_sizes, int n_in,
                              void* d_out, int out_size, void* d_ws,
                              size_t ws_size, hipStream_t stream) {
  (void)in_sizes; (void)n_in; (void)out_size; (void)d_ws; (void)ws_size;
  const float* in = (const float*)d_in[0];     // (32,3,512,512) f32
  const float* quant = (const float*)d_in[1];  // (3,8,8) f32 (only [0] used)
  float* out = (float*)d_out;                  // (32,3,512,512) f32
  // 8192 regions of 32x32 px, 8 waves (regions) per 256-thread block.
  jpeg_fused_kernel<<<1024, 256, 0, stream>>>(in, quant, out);
}